// SENetLayer_82764019793952
// MI455X (gfx1250) — compile-verified
//
#include <hip/hip_runtime.h>
#include <hip/hip_bf16.h>

// ---------------------------------------------------------------------------
// SE layer (B=32, H=W=56, C=256, R=16) for MI455X (gfx1250, wave32).
// Bandwidth-bound: ~103MB tensor read twice (2nd pass from 192MB L2) + 103MB
// written. Excitation GEMMs done with V_WMMA_F32_16X16X4_F32 (fp32-exact).
// ---------------------------------------------------------------------------

#define SE_B  32
#define SE_H  56
#define SE_W  56
#define SE_C  256
#define SE_CR 16          // C / ratio
#define SE_HW (SE_H * SE_W)   // 3136
#define SE_J  49              // hw chunks per batch: 49 * 64 = 3136

typedef float v2f __attribute__((ext_vector_type(2)));
typedef float v4f __attribute__((ext_vector_type(4)));
typedef float v8f __attribute__((ext_vector_type(8)));

// ---------------------------------------------------------------------------
// K1: partial squeeze. grid = B*J blocks x 256 threads.
// Block (b,j) reduces 64 hw positions over all 256 channels -> part[b][j][c].
// Thread t: channel-quad c4 = t%64, hw subgroup = t/64 (4-way LDS reduce).
// ---------------------------------------------------------------------------
__global__ void k_squeeze_partial(const float* __restrict__ x,
                                  float* __restrict__ part) {
    const int bj  = blockIdx.x;
    const int b   = bj / SE_J;
    const int j   = bj % SE_J;
    const int t   = threadIdx.x;
    const int c4  = t & 63;
    const int hws = t >> 6;

    const float* base = x + ((size_t)b * SE_HW + (size_t)j * 64) * SE_C + c4 * 4;
    v4f acc = {};
#pragma unroll
    for (int i = 0; i < 16; ++i) {
        const v4f v = *(const v4f*)(base + (size_t)(i * 4 + hws) * SE_C);
        acc += v;
    }

    __shared__ v4f red[256];
    red[t] = acc;
    __syncthreads();
    if (t < 64) {
        const v4f s = (red[t] + red[t + 64]) + (red[t + 128] + red[t + 192]);
        *(v4f*)(part + ((size_t)b * SE_J + j) * SE_C + c4 * 4) = s;
    }
}

// ---------------------------------------------------------------------------
// K2a: fold the J=49 partials -> s[b][c] (mean). grid = B blocks x 256 thr.
// ---------------------------------------------------------------------------
__global__ void k_reduce_mean(const float* __restrict__ part,
                              float* __restrict__ s) {
    const int b = blockIdx.x;
    const int c = threadIdx.x;
    const float* p = part + (size_t)b * SE_J * SE_C + c;
    float acc = 0.f;
#pragma unroll
    for (int j = 0; j < SE_J; ++j) acc += p[(size_t)j * SE_C];
    s[b * SE_C + c] = acc * (1.0f / (float)SE_HW);
}

// ---------------------------------------------------------------------------
// K2b: excitation MLP with fp32 WMMA. 1 block x 256 threads (8 waves).
//   GEMM1: h[32,16]  = relu(s[32,256] @ w1[256,16] + b1)   (waves 0..1)
//   GEMM2: g[32,256] = sigmoid(h[32,16] @ w2[16,256] + b2) (all 8 waves)
// WMMA fragment layouts (16x16x4 f32, wave32):
//   A: lane l, vgpr v -> M = l%16,          K = v + 2*(l/16)
//   B: lane l, vgpr v -> N = l%16,          K = v + 2*(l/16)
//   D: lane l, vgpr v -> M = v + 8*(l/16),  N = l%16
// ---------------------------------------------------------------------------
__global__ void k_excite(const float* __restrict__ s,
                         const float* __restrict__ w1,
                         const float* __restrict__ b1,
                         const float* __restrict__ w2,
                         const float* __restrict__ b2,
                         float* __restrict__ g) {
    __shared__ float s_lds[SE_B * SE_C];    // 32 KB
    __shared__ float w1_lds[SE_C * SE_CR];  // 16 KB
    __shared__ float h_lds[SE_B * SE_CR];   //  2 KB

    const int t = threadIdx.x;
    for (int i = t; i < SE_B * SE_C; i += 256)  s_lds[i]  = s[i];
    for (int i = t; i < SE_C * SE_CR; i += 256) w1_lds[i] = w1[i];
    __syncthreads();

    const int wave = t >> 5;     // 0..7 (wave32)
    const int lane = t & 31;
    const int lo   = lane & 15;  // column / M index within half
    const int hi   = lane >> 4;  // lane half select

    // ---- GEMM1: two waves, one 16-row M tile each, K = 256 in steps of 4 ----
    if (wave < 2) {
        v8f acc = {};
        const int arow = 16 * wave + lo;
#pragma unroll
        for (int kb = 0; kb < SE_C / 4; ++kb) {
            const int k0 = kb * 4 + 2 * hi;
            v2f a, bm;
            a[0]  = s_lds[arow * SE_C + k0 + 0];
            a[1]  = s_lds[arow * SE_C + k0 + 1];
            bm[0] = w1_lds[(k0 + 0) * SE_CR + lo];
            bm[1] = w1_lds[(k0 + 1) * SE_CR + lo];
            acc = __builtin_amdgcn_wmma_f32_16x16x4_f32(
                false, a, false, bm, (short)0, acc, false, false);
        }
        const float bias = b1[lo];
#pragma unroll
        for (int v = 0; v < 8; ++v) {
            const int row = 16 * wave + v + 8 * hi;
            const float hv = acc[v] + bias;
            h_lds[row * SE_CR + lo] = hv > 0.f ? hv : 0.f;
        }
    }
    __syncthreads();

    // ---- GEMM2: 32 output tiles (2 M x 16 N), K = 16 in steps of 4 ----
    for (int tile = wave; tile < 32; tile += 8) {
        const int m = tile >> 4;   // 0..1
        const int n = tile & 15;   // 0..15
        v8f acc = {};
#pragma unroll
        for (int kb = 0; kb < SE_CR / 4; ++kb) {
            const int k0 = kb * 4 + 2 * hi;
            v2f a, bm;
            a[0]  = h_lds[(16 * m + lo) * SE_CR + k0 + 0];
            a[1]  = h_lds[(16 * m + lo) * SE_CR + k0 + 1];
            bm[0] = w2[(k0 + 0) * SE_C + 16 * n + lo];
            bm[1] = w2[(k0 + 1) * SE_C + 16 * n + lo];
            acc = __builtin_amdgcn_wmma_f32_16x16x4_f32(
                false, a, false, bm, (short)0, acc, false, false);
        }
        const float bias = b2[16 * n + lo];
#pragma unroll
        for (int v = 0; v < 8; ++v) {
            const int row = 16 * m + v + 8 * hi;
            const float gv = acc[v] + bias;
            g[row * SE_C + 16 * n + lo] = 1.0f / (1.0f + expf(-gv));
        }
    }
}

// ---------------------------------------------------------------------------
// K3: out = x * g (broadcast over hw). grid = B*J blocks x 256 threads.
// v4f loads of x (should hit L2 from pass 1); nontemporal b128 stores of
// out so the 103MB output stream doesn't evict x from L2.
// ---------------------------------------------------------------------------
__global__ void k_scale(const float* __restrict__ x,
                        const float* __restrict__ g,
                        float* __restrict__ out) {
    const int bj  = blockIdx.x;
    const int b   = bj / SE_J;
    const int j   = bj % SE_J;
    const int t   = threadIdx.x;
    const int c4  = t & 63;
    const int hws = t >> 6;

    const v4f gv = *(const v4f*)(g + b * SE_C + c4 * 4);
    const size_t base = ((size_t)b * SE_HW + (size_t)j * 64) * SE_C + c4 * 4;
#pragma unroll
    for (int i = 0; i < 16; ++i) {
        const size_t off = base + (size_t)(i * 4 + hws) * SE_C;
        const v4f v = *(const v4f*)(x + off) * gv;
        __builtin_nontemporal_store(v, (v4f*)(out + off));
    }
}

// ---------------------------------------------------------------------------
extern "C" void kernel_launch(void* const* d_in, const int* in_sizes, int n_in,
                              void* d_out, int out_size, void* d_ws, size_t ws_size,
                              hipStream_t stream) {
    (void)in_sizes; (void)n_in; (void)out_size; (void)ws_size;
    const float* x  = (const float*)d_in[0];
    const float* w1 = (const float*)d_in[1];
    const float* b1 = (const float*)d_in[2];
    const float* w2 = (const float*)d_in[3];
    const float* b2 = (const float*)d_in[4];
    float* out = (float*)d_out;

    // Big partial-sum scratch lives at the start of d_out (1.6MB, fully
    // overwritten by k_scale at the end). ws only needs 64KB for s and g.
    float* part = out;
    float* s    = (float*)d_ws;
    float* g    = s + SE_B * SE_C;

    k_squeeze_partial<<<SE_B * SE_J, 256, 0, stream>>>(x, part);
    k_reduce_mean   <<<SE_B,         256, 0, stream>>>(part, s);
    k_excite        <<<1,            256, 0, stream>>>(s, w1, b1, w2, b2, g);
    k_scale         <<<SE_B * SE_J,  256, 0, stream>>>(x, g, out);
}